// AdaptivePromptGenerator_81484119540137
// MI455X (gfx1250) — compile-verified
//
#include <hip/hip_runtime.h>
#include <hip/hip_bf16.h>
#include <cstdint>

typedef __attribute__((ext_vector_type(2))) float v2f;
typedef __attribute__((ext_vector_type(8))) float v8f;

#define HH 1024
#define WW 1024
#define NBC 32      // 16 batches * 2 channels
#define CAP 4096    // max peak candidates per (b,c)

// Normalized gaussian taps, sigma=2, ksize=5
__device__ __forceinline__ float gw(int d) {
  float r = 0.0f;
  r = (d == 0 || d == 4) ? 0.15247190f : r;
  r = (d == 1 || d == 3) ? 0.22184130f : r;
  r = (d == 2)           ? 0.25137904f : r;
  return r;
}

__global__ void init_kernel(int* __restrict__ boxi, int* __restrict__ counters) {
  int t = threadIdx.x;
  if (t < NBC) {
    counters[t] = 0;
    boxi[t * 4 + 0] = WW;   // min col
    boxi[t * 4 + 1] = HH;   // min row
    boxi[t * 4 + 2] = -1;   // max col
    boxi[t * 4 + 3] = -1;   // max row
  }
}

// Pass 1: separable gaussian blur. Vertical tap via VALU into LDS,
// horizontal tap as out(16x16) = T(16x20) x G(20x16) via 5x WMMA f32 16x16x4.
// Fused: per-(b,c) thresholded bounding-box reduction (wave shuffle + atomics).
__global__ __launch_bounds__(256) void blur_kernel(const float* __restrict__ in,
                                                   float* __restrict__ sm,
                                                   int* __restrict__ boxi) {
  __shared__ float patch[8][400];   // 20x20 input patch per wave
  __shared__ float tbuf[8][320];    // 16x20 vertically-blurred per wave
  const int wave = threadIdx.x >> 5;
  const int lane = threadIdx.x & 31;
  const int bc = blockIdx.x >> 9;          // / 512 blocks per (b,c)
  const int bt = blockIdx.x & 511;
  const int tileRow = bt >> 3;             // 0..63
  const int tileCol = ((bt & 7) << 3) + wave;  // 0..63
  const int r0 = tileRow * 16, c0 = tileCol * 16;
  const size_t base = (size_t)bc * ((size_t)HH * WW);

  // Stage 20x20 input patch (zero-padded) into LDS
  for (int idx = lane; idx < 400; idx += 32) {
    int r = idx / 20, c = idx - r * 20;
    int gr = r0 - 2 + r, gc = c0 - 2 + c;
    float v = 0.0f;
    if ((unsigned)gr < (unsigned)HH && (unsigned)gc < (unsigned)WW)
      v = in[base + (size_t)gr * WW + gc];
    patch[wave][idx] = v;
  }
  __syncthreads();

  // Vertical 5-tap blur -> T (16 rows x 20 cols)
  for (int idx = lane; idx < 320; idx += 32) {
    int r = idx / 20, c = idx - r * 20;
    const float* p = &patch[wave][r * 20 + c];
    tbuf[wave][idx] = 0.15247190f * (p[0] + p[80]) +
                      0.22184130f * (p[20] + p[60]) +
                      0.25137904f * p[40];
  }
  __syncthreads();

  // Horizontal blur as WMMA: D = sum_k A_k(16x4) x B_k(4x16)
  // A layout (f32 16x4): lanes 0-15 M=0..15 {K0,K1}, lanes 16-31 M=0..15 {K2,K3}
  const int hi   = (lane >> 4) & 1;
  const int m    = lane & 15;     // row M (A) / col N (B,D)
  const int koff = hi * 2;
  v8f acc = {0.f, 0.f, 0.f, 0.f, 0.f, 0.f, 0.f, 0.f};
#pragma unroll
  for (int k = 0; k < 5; ++k) {
    v2f a, b;
    a.x = tbuf[wave][m * 20 + 4 * k + koff];
    a.y = tbuf[wave][m * 20 + 4 * k + koff + 1];
    int j0 = 4 * k + koff;                 // B row K index (global 0..19)
    b.x = gw(j0 - m);                      // Gh[j][n] = g[j-n] if in [0,5)
    b.y = gw(j0 + 1 - m);
    acc = __builtin_amdgcn_wmma_f32_16x16x4_f32(
        false, a, false, b, (short)0, acc, false, false);
  }

  // D layout: VGPR v -> row r0+v (lanes 0-15) / r0+8+v (lanes 16-31), col c0+m
  const int col = c0 + m;
  const int rbase = r0 + hi * 8;
  int rmin = HH, rmax = -1;
#pragma unroll
  for (int v = 0; v < 8; ++v) {
    int row = rbase + v;
    float x = acc[v];
    sm[base + (size_t)row * WW + col] = x;
    if (x >= 0.4f) { rmin = min(rmin, row); rmax = max(rmax, row); }
  }
  int cmin = (rmax >= 0) ? col : WW;
  int cmax = (rmax >= 0) ? col : -1;
#pragma unroll
  for (int off = 16; off > 0; off >>= 1) {
    rmin = min(rmin, __shfl_xor(rmin, off));
    cmin = min(cmin, __shfl_xor(cmin, off));
    rmax = max(rmax, __shfl_xor(rmax, off));
    cmax = max(cmax, __shfl_xor(cmax, off));
  }
  if (lane == 0 && rmax >= 0) {
    atomicMin(&boxi[bc * 4 + 0], cmin);
    atomicMin(&boxi[bc * 4 + 1], rmin);
    atomicMax(&boxi[bc * 4 + 2], cmax);
    atomicMax(&boxi[bc * 4 + 3], rmax);
  }
}

// Pass 2: 31x31 max-pool via van Herk-Gil-Werman (block = window = 31, so
// window max = max(suffix[j], prefix[j+30]) over 31-aligned blocks), then
// peak detect. Tile: 64 out cols x 32 out rows per block; sm is L2-resident.
__global__ __launch_bounds__(256) void peak_kernel(const float* __restrict__ sm,
                                                   int* __restrict__ counters,
                                                   uint32_t* __restrict__ cand) {
  __shared__ float smem[15376];            // 61.5 KB, aliased across stages
  float* s_suf = smem;                     // [62][124] row suffix-max
  float* s_pre = smem + 62 * 124;          // [62][124] row prefix-max
  const int tid = threadIdx.x;
  const int c0 = blockIdx.x * 64;
  const int r0 = blockIdx.y * 32;
  const int bc = blockIdx.z;
  const size_t base = (size_t)bc * ((size_t)HH * WW);
  const float NEG = -3.0e38f;

  // Stage A: one thread per (tile row, 31-col block): 62*4 = 248 segments.
  // Load 31 contiguous floats (zero... -inf padded), build prefix/suffix in regs.
  if (tid < 248) {
    const int r = tid >> 2;          // 0..61  (global row r0-15+r)
    const int blk = tid & 3;         // 0..3   (tile cols 31*blk .. 31*blk+30)
    const int gr = r0 - 15 + r;
    float x[31];
#pragma unroll
    for (int t = 0; t < 31; ++t) {
      int gc = c0 - 15 + blk * 31 + t;
      float v = NEG;
      if ((unsigned)gr < (unsigned)HH && (unsigned)gc < (unsigned)WW)
        v = sm[base + (size_t)gr * WW + gc];
      x[t] = v;
    }
    float pre[31];
    pre[0] = x[0];
#pragma unroll
    for (int t = 1; t < 31; ++t) pre[t] = fmaxf(pre[t - 1], x[t]);
#pragma unroll
    for (int t = 29; t >= 0; --t) x[t] = fmaxf(x[t], x[t + 1]);   // x -> suffix
    float* su = &s_suf[r * 124 + blk * 31];
    float* pr = &s_pre[r * 124 + blk * 31];
#pragma unroll
    for (int t = 0; t < 31; ++t) { su[t] = x[t]; pr[t] = pre[t]; }
  }
  __syncthreads();

  // Stage B: one thread per (tile col, 31-row block): 64*2 = 128 segments.
  // Row-max rm[r][j] = max(suf[r][j], pre[r][j+30]) formed on the fly,
  // column van Herk prefix/suffix built in registers.
  float cp[31], cs[31];
  const int j = tid & 63;
  const int rb = (tid >> 6) * 31;    // 0 or 31 (rows 0..61 = exactly 2 blocks)
  if (tid < 128) {
#pragma unroll
    for (int t = 0; t < 31; ++t) {
      int r = rb + t;
      cs[t] = fmaxf(s_suf[r * 124 + j], s_pre[r * 124 + j + 30]);  // rm value
    }
    cp[0] = cs[0];
#pragma unroll
    for (int t = 1; t < 31; ++t) cp[t] = fmaxf(cp[t - 1], cs[t]);
#pragma unroll
    for (int t = 29; t >= 0; --t) cs[t] = fmaxf(cs[t], cs[t + 1]); // col suffix
  }
  __syncthreads();   // all reads of s_suf/s_pre complete before aliasing

  float* s_cs = smem;                // [62][64] column suffix-max
  float* s_cp = smem + 62 * 64;      // [62][64] column prefix-max
  if (tid < 128) {
#pragma unroll
    for (int t = 0; t < 31; ++t) {
      s_cs[(rb + t) * 64 + j] = cs[t];
      s_cp[(rb + t) * 64 + j] = cp[t];
    }
  }
  __syncthreads();

  // Final: 31x31 window max = max(colsuf[i][j], colpre[i+30][j]); peak test.
  for (int idx = tid; idx < 32 * 64; idx += 256) {
    int i = idx >> 6, jj = idx & 63;
    float mp = fmaxf(s_cs[i * 64 + jj], s_cp[(i + 30) * 64 + jj]);
    float v = sm[base + (size_t)(r0 + i) * WW + (c0 + jj)];
    if (v == mp && v >= 0.5f) {
      int slot = atomicAdd(&counters[bc], 1);
      if (slot < CAP) {
        uint32_t* p = &cand[((size_t)bc * CAP + slot) * 2];
        p[0] = __float_as_uint(v);
        p[1] = (uint32_t)((r0 + i) * WW + (c0 + jj));
      }
    }
  }
}

// Pass 3: per-(b,c) box margin/clip + deterministic top-3 (score desc, idx asc).
__global__ __launch_bounds__(256) void final_kernel(const int* __restrict__ boxi,
                                                    const int* __restrict__ counters,
                                                    const uint32_t* __restrict__ cand,
                                                    float* __restrict__ out) {
  __shared__ float rv[256];
  __shared__ int   ri[256];
  __shared__ float bestv[3];
  __shared__ int   besti[3];
  const int tid = threadIdx.x;
  const int bc = blockIdx.x;
  int cnt = counters[bc]; if (cnt > CAP) cnt = CAP;
  const uint32_t* cp = &cand[(size_t)bc * CAP * 2];

  for (int k = 0; k < 3; ++k) {
    float bv = -2.0f; int bi = 0x7FFFFFFF;
    for (int c = tid; c < cnt; c += 256) {
      float v = __uint_as_float(cp[c * 2 + 0]);
      int   id = (int)cp[c * 2 + 1];
      bool dup = false;
      for (int q = 0; q < k; ++q) dup = dup || (bestv[q] > -1.5f && besti[q] == id);
      if (!dup && (v > bv || (v == bv && id < bi))) { bv = v; bi = id; }
    }
    rv[tid] = bv; ri[tid] = bi;
    __syncthreads();
    for (int s = 128; s > 0; s >>= 1) {
      if (tid < s) {
        float v2 = rv[tid + s]; int i2 = ri[tid + s];
        if (v2 > rv[tid] || (v2 == rv[tid] && i2 < ri[tid])) { rv[tid] = v2; ri[tid] = i2; }
      }
      __syncthreads();
    }
    if (tid == 0) { bestv[k] = rv[0]; besti[k] = ri[0]; }
    __syncthreads();
  }

  if (tid == 0) {
    // boxes
    int xi0 = boxi[bc*4+0], yi0 = boxi[bc*4+1], xi1 = boxi[bc*4+2], yi1 = boxi[bc*4+3];
    bool any = (xi1 >= 0);
    float x0 = any ? (float)xi0 : 0.f;
    float y0 = any ? (float)yi0 : 0.f;
    float x1 = any ? (float)xi1 : (float)(WW - 1);
    float y1 = any ? (float)yi1 : (float)(HH - 1);
    float mw = (x1 - x0) * 0.1f, mh = (y1 - y0) * 0.1f;
    x0 = fminf(fmaxf(x0 - mw, 0.f), (float)(WW - 1));
    x1 = fminf(fmaxf(x1 + mw, 0.f), (float)(WW - 1));
    y0 = fminf(fmaxf(y0 - mh, 0.f), (float)(HH - 1));
    y1 = fminf(fmaxf(y1 + mh, 0.f), (float)(HH - 1));
    out[bc*4+0] = x0; out[bc*4+1] = y0; out[bc*4+2] = x1; out[bc*4+3] = y1;

    // points / labels / scores (top_k semantics incl. -1 fillers at lowest free idx)
    float* pts = out + 128;
    float* lbl = out + 320;
    float* scr = out + 416;
    int fptr = 0;
    for (int k = 0; k < 3; ++k) {
      float v; int id;
      if (bestv[k] > -1.5f) { v = bestv[k]; id = besti[k]; }
      else {
        bool hit = true;
        while (hit) {
          hit = false;
          for (int q = 0; q < 3; ++q)
            if (bestv[q] > -1.5f && besti[q] == fptr) hit = true;
          if (hit) fptr++;
        }
        id = fptr++; v = -1.0f;
      }
      pts[(bc*3 + k)*2 + 0] = (float)(id % WW);
      pts[(bc*3 + k)*2 + 1] = (float)(id / WW);
      lbl[bc*3 + k] = 1.0f;
      scr[bc*3 + k] = v;
    }
  }
}

extern "C" void kernel_launch(void* const* d_in, const int* in_sizes, int n_in,
                              void* d_out, int out_size, void* d_ws, size_t ws_size,
                              hipStream_t stream) {
  (void)in_sizes; (void)n_in; (void)out_size; (void)ws_size;
  const float* heat = (const float*)d_in[0];
  float* out = (float*)d_out;

  // workspace layout
  float* sm = (float*)d_ws;                                   // 128 MiB blurred field
  const size_t SM_BYTES = (size_t)NBC * HH * WW * sizeof(float);
  int* boxi = (int*)((char*)d_ws + SM_BYTES);                 // 128 ints
  int* counters = boxi + 4 * NBC;                             // 32 ints
  uint32_t* cand = (uint32_t*)(counters + NBC);               // 32*CAP*2 u32

  init_kernel<<<1, 64, 0, stream>>>(boxi, counters);
  blur_kernel<<<NBC * 512, 256, 0, stream>>>(heat, sm, boxi);
  peak_kernel<<<dim3(16, 32, NBC), 256, 0, stream>>>(sm, counters, cand);
  final_kernel<<<NBC, 256, 0, stream>>>(boxi, counters, cand, out);
}